// HTMask_17944373363369
// MI455X (gfx1250) — compile-verified
//
#include <hip/hip_runtime.h>
#include <math.h>

// ---------------------------------------------------------------------------
// Problem constants (from the reference)
// ---------------------------------------------------------------------------
#define BSZ   8
#define CIN   128
#define COUT  32
#define HH    128
#define WW    128
#define AA    180
#define RR    184
#define K1DIM (CIN * 9)        // 1152 : total K of the implicit 3x3 GEMM
#define NPIX  (HH * WW)        // 16384

typedef __attribute__((ext_vector_type(2))) float v2f;
typedef __attribute__((ext_vector_type(8))) float v8f;

// D(16x16,f32) = A(16x4,f32) x B(4x16,f32) + C  -- full-precision CDNA5 WMMA.
__device__ __forceinline__ v8f wmma_f32(v2f a, v2f b, v8f c) {
    return __builtin_amdgcn_wmma_f32_16x16x4_f32(
        false, a, false, b, (short)0, c, false, false);
}

// irho = sqrt(H*H + W*W) / R  (f64 on host side of reference)
#define IRHO_F ((float)(181.01933598375618 / 184.0))
#define PI_OVER_A ((float)(3.14159265358979323846 / 180.0))

// Staged x tile is ci-minor: smem[col][ci], col in 0..129 (1-col zero halo on
// each side), ci-stride 68 (=64+4 pad). B operand pairs (ci, ci+1) are then
// adjacent -> one aligned ds_load_b64 per WMMA, and the 32-lane access
// pattern (bank step 4/lane, 4 words per lane-pair) covers all 64 LDS banks
// exactly once -> conflict-free.
#define CSTR 68

// ---------------------------------------------------------------------------
// Kernel 1: fused conv3x3(128->128) + BN + ReLU + conv1x1(128->32) + bias,
// plus per-pixel channel-sum for the Hough stage.
//
// One block = one (batch, image-row). 8 waves; wave w owns conv1 output
// channels [16w,16w+16). The 3x3 conv is decomposed into 9 shifted GEMMs
// (outer loop over kh,kw; K = ci). Inner K-step: 2 global b32 (A weights),
// 8 ds_load_b64 (B), 8 WMMA. No bounds checks, no divides, no divergence.
// ---------------------------------------------------------------------------
__global__ __launch_bounds__(256)
void k1_conv_bn_conv(const float* __restrict__ x,
                     const float* __restrict__ w1,
                     const float* __restrict__ gamma,
                     const float* __restrict__ beta,
                     const float* __restrict__ mean,
                     const float* __restrict__ var,
                     const float* __restrict__ w2,
                     const float* __restrict__ b2,
                     float* __restrict__ out,
                     float* __restrict__ s_sum)
{
    // 64 KB union: x staging tile (130 x 68 = 8840 floats) during GEMM1,
    // then the full h tile (128 ci2 x 128 pix) for GEMM2.
    __shared__ float smem[CIN * WW];

    const int y       = blockIdx.x;
    const int bb      = blockIdx.y;
    const int tid     = threadIdx.x;
    const int wv      = tid >> 5;        // wave id 0..7 (wave32)
    const int lane    = tid & 31;
    const int lanelo  = lane & 15;
    const bool hihalf = lane >= 16;
    const int dK      = hihalf ? 2 : 0;  // K-offset of this lane half
    const int co_base = wv * 16;

    v8f acc[8];
    #pragma unroll
    for (int n = 0; n < 8; ++n) acc[n] = (v8f)0.0f;

    const float* xb  = x  + (size_t)bb * CIN * NPIX;
    const float* w1c = w1 + (size_t)(co_base + lanelo) * K1DIM;

    // ---- GEMM1: 9 shifted GEMMs over (kh, kw), K = ci in halves of 64 ----
    for (int kh = 0; kh < 3; ++kh) {
        const int row = y + kh - 1;
        if ((unsigned)row >= (unsigned)HH) continue;   // block-uniform skip

        for (int cih = 0; cih < 2; ++cih) {
            const int ci0 = cih * 64;
            __syncthreads();   // previous slab's LDS reads complete
            // Stage 64 ci-rows, ci-minor, with 1-col zero halo each side.
            for (int f = tid; f < 64 * 130; f += 256) {
                int ci  = f / 130;
                int cc  = f - ci * 130;      // 0..129 ; col = cc-1
                int col = cc - 1;
                float v = ((unsigned)col < (unsigned)WW)
                        ? xb[((size_t)(ci0 + ci) * HH + row) * WW + col]
                        : 0.0f;
                smem[cc * CSTR + ci] = v;
            }
            __syncthreads();

            #pragma unroll
            for (int kw = 0; kw < 3; ++kw) {
                // A element pointer for this lane: w1[co][ci0+ci4+dK][kh][kw]
                const float* w1a = w1c + (size_t)(ci0 + dK) * 9 + kh * 3 + kw;
                // B base: column (pix + kw) (halo +1 and dx=kw-1 combine), this
                // lane half's first K row.
                const float* bp = smem + (size_t)(lanelo + kw) * CSTR + dK;
                #pragma unroll 2
                for (int ci4 = 0; ci4 < 64; ci4 += 4) {
                    v2f av;
                    av.x = w1a[ci4 * 9];          // K = ci4+dK
                    av.y = w1a[ci4 * 9 + 9];      // K = ci4+dK+1
                    #pragma unroll
                    for (int n = 0; n < 8; ++n) {
                        // one aligned ds_load_b64: rows (ci4+dK, ci4+dK+1)
                        v2f bv = *(const v2f*)(bp + (size_t)n * 16 * CSTR + ci4);
                        acc[n] = wmma_f32(av, bv, acc[n]);
                    }
                }
            }
        }
    }

    // ---- BN + ReLU epilogue -> LDS h tile (K-major, stride WW) ----
    __syncthreads();   // all staging reads done; reuse smem for h
    #pragma unroll
    for (int r = 0; r < 8; ++r) {
        // D layout: VGPR r, lanes0-15 -> M=r, N=lane ; lanes16-31 -> M=r+8
        int co   = co_base + r + (hihalf ? 8 : 0);
        float iv = gamma[co] / sqrtf(var[co] + 1e-5f);
        float bc = beta[co] - mean[co] * iv;
        #pragma unroll
        for (int n = 0; n < 8; ++n) {
            float hv = fmaxf(acc[n][r] * iv + bc, 0.0f);
            smem[co * WW + n * 16 + lanelo] = hv;
        }
    }
    __syncthreads();

    // ---- GEMM2: 1x1 conv, M=32 (two 16-tiles), wave w owns pixel tile w ----
    v8f acc2[2];
    acc2[0] = (v8f)0.0f;
    acc2[1] = (v8f)0.0f;
    const int pix = wv * 16 + lanelo;
    const float* w2r0 = w2 + (size_t)lanelo * CIN;          // co2 0..15
    const float* w2r1 = w2 + (size_t)(16 + lanelo) * CIN;   // co2 16..31

    for (int kb = 0; kb < CIN; kb += 4) {
        int ka = kb + dK;
        v2f a0, a1, bv;
        a0.x = w2r0[ka];  a0.y = w2r0[ka + 1];
        a1.x = w2r1[ka];  a1.y = w2r1[ka + 1];
        bv.x = smem[ka * WW + pix];
        bv.y = smem[(ka + 1) * WW + pix];
        acc2[0] = wmma_f32(a0, bv, acc2[0]);
        acc2[1] = wmma_f32(a1, bv, acc2[1]);
    }

    // ---- write out + accumulate channel sum s[b,p] (for Hough stage) ----
    float psum = 0.0f;
    #pragma unroll
    for (int mt = 0; mt < 2; ++mt) {
        #pragma unroll
        for (int r = 0; r < 8; ++r) {
            int co2  = mt * 16 + r + (hihalf ? 8 : 0);
            float v  = acc2[mt][r] + b2[co2];
            out[((size_t)bb * COUT + co2) * NPIX + y * WW + pix] = v;
            psum += v;
        }
    }
    atomicAdd(&s_sum[(size_t)bb * NPIX + y * WW + pix], psum);
}

// ---------------------------------------------------------------------------
// Kernel 2: Hough binning. One block per (angle, batch); 184 bins in LDS.
// hm[b,a,r] = (1/COUT) * sum_{p : r(a,p)==r} s[b,p]
// ---------------------------------------------------------------------------
__global__ __launch_bounds__(256)
void k2_hough(const float* __restrict__ s_sum, float* __restrict__ hm)
{
    __shared__ float bins[RR];
    const int a   = blockIdx.x;
    const int bb  = blockIdx.y;
    const int tid = threadIdx.x;

    if (tid < RR) bins[tid] = 0.0f;
    __syncthreads();

    const float theta = (float)a * PI_OVER_A;
    const float ct = cosf(theta), st = sinf(theta);
    const float* sb = s_sum + (size_t)bb * NPIX;

    for (int p = tid; p < NPIX; p += 256) {
        int i = p >> 7, j = p & 127;
        float xc  = (float)j - 63.5f;
        float yc  = (float)i - 63.5f;
        float rho = xc * ct + yc * st;
        int r = (int)rintf(rho / IRHO_F) + RR / 2;   // round-half-even == jnp.round
        r = min(max(r, 0), RR - 1);
        atomicAdd(&bins[r], sb[p]);
    }
    __syncthreads();
    if (tid < RR)
        hm[((size_t)bb * AA + a) * RR + tid] = bins[tid] * (1.0f / COUT);
}

// ---------------------------------------------------------------------------
// Kernel 3: gather + sigmoid mask, broadcast over COUT channels.
// hm[b] staged through LDS in 3 chunks of 60 angles (static LDS < 64KB).
// One thread per pixel.
// ---------------------------------------------------------------------------
#define ACHUNK 60
__global__ __launch_bounds__(256)
void k3_mask(const float* __restrict__ hm, float* __restrict__ mask_out)
{
    __shared__ float hml[ACHUNK * RR];   // 44160 B
    __shared__ float ctab[AA], stab[AA];

    const int bb  = blockIdx.y;
    const int tid = threadIdx.x;
    const int p   = blockIdx.x * 256 + tid;

    for (int a = tid; a < AA; a += 256) {
        float theta = (float)a * PI_OVER_A;
        ctab[a] = cosf(theta);
        stab[a] = sinf(theta);
    }

    const int i = p >> 7, j = p & 127;
    const float xc = (float)j - 63.5f;
    const float yc = (float)i - 63.5f;

    float accv = 0.0f;
    for (int c0 = 0; c0 < AA; c0 += ACHUNK) {
        __syncthreads();   // tables ready / previous chunk consumed
        const float* src = hm + ((size_t)bb * AA + c0) * RR;
        for (int t = tid; t < ACHUNK * RR; t += 256) hml[t] = src[t];
        __syncthreads();
        #pragma unroll 4
        for (int a = 0; a < ACHUNK; ++a) {
            float rho = xc * ctab[c0 + a] + yc * stab[c0 + a];
            int r = (int)rintf(rho / IRHO_F) + RR / 2;
            r = min(max(r, 0), RR - 1);
            accv += hml[a * RR + r];
        }
    }

    float m = 1.0f / (1.0f + expf(-accv));
    float* mb = mask_out + (size_t)bb * COUT * NPIX;
    #pragma unroll
    for (int c = 0; c < COUT; ++c)
        mb[(size_t)c * NPIX + p] = m;
}

// ---------------------------------------------------------------------------
// Launch: out tuple is (mask, out) concatenated flat. Workspace holds the
// per-pixel channel sums s (512 KB) and the Hough table hm (~1 MB).
// ---------------------------------------------------------------------------
extern "C" void kernel_launch(void* const* d_in, const int* in_sizes, int n_in,
                              void* d_out, int out_size, void* d_ws, size_t ws_size,
                              hipStream_t stream)
{
    const float* x     = (const float*)d_in[0];
    const float* w1    = (const float*)d_in[1];
    const float* gamma = (const float*)d_in[2];
    const float* beta  = (const float*)d_in[3];
    const float* mean  = (const float*)d_in[4];
    const float* var   = (const float*)d_in[5];
    const float* w2    = (const float*)d_in[6];
    const float* b2    = (const float*)d_in[7];

    float* mask = (float*)d_out;                                // B*COUT*H*W
    float* out2 = (float*)d_out + (size_t)BSZ * COUT * NPIX;    // B*COUT*H*W

    float* s_sum = (float*)d_ws;                    // B*H*W floats
    float* hm    = s_sum + (size_t)BSZ * NPIX;      // B*A*R floats

    hipMemsetAsync(s_sum, 0, sizeof(float) * (size_t)BSZ * NPIX, stream);

    k1_conv_bn_conv<<<dim3(HH, BSZ), 256, 0, stream>>>(
        x, w1, gamma, beta, mean, var, w2, b2, out2, s_sum);

    k2_hough<<<dim3(AA, BSZ), 256, 0, stream>>>(s_sum, hm);

    k3_mask<<<dim3(NPIX / 256, BSZ), 256, 0, stream>>>(hm, mask);
}